// Pair_to_MSA_65377992179734
// MI455X (gfx1250) — compile-verified
//
#include <hip/hip_runtime.h>
#include <hip/hip_bf16.h>
#include <cfloat>

// Problem constants (from reference)
#define RR 256
#define LL 384
#define IND 384      // IN_DIM
#define CEMB 128     // N_EMB_2D
#define NH 12        // N_HEAD
#define OUTD 384     // OUT_DIM
#define HD 32        // head dim

typedef __attribute__((ext_vector_type(2))) float v2f;
typedef __attribute__((ext_vector_type(8))) float v8f;
typedef __attribute__((ext_vector_type(4))) unsigned int u32x4;
typedef __attribute__((ext_vector_type(8))) int i32x8;
typedef __attribute__((ext_vector_type(4))) int i32x4;

__device__ __forceinline__ float waveMax(float v) {
#pragma unroll
  for (int off = 16; off > 0; off >>= 1) v = fmaxf(v, __shfl_xor(v, off, 32));
  return v;
}
__device__ __forceinline__ float waveSum(float v) {
#pragma unroll
  for (int off = 16; off > 0; off >>= 1) v += __shfl_xor(v, off, 32);
  return v;
}

// ---------------------------------------------------------------------------
// TDM: issue a 2D tensor tile load (fp32 elements) Global -> LDS.
// D# layout per CDNA5 ISA ch.8: group0 = {count/type, lds_addr, global_addr},
// group1 = {data_size, tensor dims, tile dims, dim0 stride}.  The tile is
// written row-major into LDS: [tile_d1][tile_d0] floats.
// This toolchain's builtin takes 6 args (extra trailing int32x8 group).
// ---------------------------------------------------------------------------
__device__ __forceinline__ void tdm_load_2d_f32(unsigned lds_addr,
                                                const float* gptr,
                                                unsigned tensor_d0, unsigned tensor_d1,
                                                unsigned tile_d0, unsigned tile_d1,
                                                unsigned stride0 /* elements */) {
  unsigned long long ga = (unsigned long long)(uintptr_t)gptr;
  u32x4 g0 = {
      1u,                                             // count=1, user mode
      lds_addr,                                       // LDS byte address
      (unsigned)(ga & 0xffffffffull),                 // global_addr[31:0]
      (unsigned)((ga >> 32) & 0x1ffffffull) | (2u << 30)  // addr[56:32] | type=2
  };
  i32x8 g1 = {
      (int)(2u << 16),                                // data_size=4B, no mask/pad
      (int)((tensor_d0 & 0xffffu) << 16),             // tensor_dim0[15:0]
      (int)(((tensor_d0 >> 16) & 0xffffu) | ((tensor_d1 & 0xffffu) << 16)),
      (int)(((tensor_d1 >> 16) & 0xffffu) | ((tile_d0 & 0xffffu) << 16)),
      (int)(tile_d1 & 0xffffu),                       // tile_dim1, tile_dim2=0
      (int)stride0,                                   // tensor_dim0_stride[31:0]
      0, 0
  };
  i32x4 z4 = {0, 0, 0, 0};
  i32x8 z8 = {0, 0, 0, 0, 0, 0, 0, 0};
  __builtin_amdgcn_tensor_load_to_lds(g0, g1, z4, z4, z8, 0);
}

// ---------------------------------------------------------------------------
// Kernel 1: logits[h,i,j] = scale * sum_c pair[i,j,c] * W_l[h,c]  (WMMA, K=128)
// then masked softmax over j, write attn[h,i,j] to workspace.
// Grid: LL blocks (one per i). Block: 384 threads = 12 waves.
// ---------------------------------------------------------------------------
__global__ void attn_logits_softmax_kernel(const float* __restrict__ pair,
                                           const float* __restrict__ Wl,
                                           const int*   __restrict__ mask,
                                           float*       __restrict__ attn) {
  __shared__ float sWl[16 * CEMB];     // 8 KB, rows 12..15 zero
  __shared__ float slog[NH][LL];       // 18 KB

  const int i    = blockIdx.x;
  const int tid  = threadIdx.x;
  const int wave = tid >> 5;
  const int lane = tid & 31;

  for (int t = tid; t < 16 * CEMB; t += blockDim.x)
    sWl[t] = (t < NH * CEMB) ? Wl[t] : 0.0f;
  __syncthreads();

  const float scale = 0.051031036307982884f;  // 1/sqrt(384)
  const int m  = lane & 15;
  const int kb = (lane >> 4) << 1;
  const int hi = (lane >> 4) ? 8 : 0;

  const float* bbase = &sWl[m * CEMB + kb];

#pragma unroll
  for (int t = 0; t < 2; ++t) {
    const int j0 = (wave * 2 + t) * 16;
    const float* prow = pair + ((size_t)i * LL + (j0 + m)) * CEMB + kb;
    v8f c = {};
#pragma unroll
    for (int c0 = 0; c0 < CEMB; c0 += 4) {
      v2f a = *(const v2f*)(prow + c0);
      v2f b = *(const v2f*)(bbase + c0);
      c = __builtin_amdgcn_wmma_f32_16x16x4_f32(false, a, false, b,
                                                (short)0, c, false, false);
    }
#pragma unroll
    for (int v = 0; v < 8; ++v) {
      const int jj = j0 + v + hi;
      if (m < NH) slog[m][jj] = c[v] * scale;
    }
  }
  __syncthreads();

  const int h = wave;
  const int maskI = mask[i];
  float mx = -FLT_MAX;
  for (int j = lane; j < LL; j += 32) {
    float l = slog[h][j];
    if (maskI == 0 || mask[j] == 0) l = -FLT_MAX;
    slog[h][j] = l;
    mx = fmaxf(mx, l);
  }
  mx = waveMax(mx);
  float sum = 0.0f;
  for (int j = lane; j < LL; j += 32) {
    float e = __expf(slog[h][j] - mx);
    slog[h][j] = e;
    sum += e;
  }
  sum = waveSum(sum);
  const float inv = 1.0f / sum;
  float* arow = attn + ((size_t)h * LL + i) * LL;
  for (int j = lane; j < LL; j += 32) arow[j] = slog[h][j] * inv;
}

// ---------------------------------------------------------------------------
// Kernel 2: V[row, o] = sum_e msa[row, e] * W_v[o, e]   rows = R*L = 98304
// Block = 128 rows x 64-col strip (8 waves, each 16 rows x 64 cols).
// W_v panel (64 rows x 96 K) staged in LDS via TDM, double buffered.
// Grid: (98304/128)*(384/64) = 4608 blocks of 256 threads.
// ---------------------------------------------------------------------------
__global__ void v_gemm_kernel(const float* __restrict__ msa,
                              const float* __restrict__ Wv,
                              float*       __restrict__ V) {
  __shared__ float sB[2][64][96];      // 48 KB double buffer

  const int tid  = threadIdx.x;
  const int wave = tid >> 5;
  const int lane = tid & 31;

  const int NT   = OUTD / 64;          // 6
  const int ct   = blockIdx.x % NT;
  const int rt   = blockIdx.x / NT;
  const int col0 = ct * 64;
  const int row0 = rt * 128 + wave * 16;

  const int m  = lane & 15;
  const int kb = (lane >> 4) << 1;
  const int hi = (lane >> 4) ? 8 : 0;

  const float* arow = msa + (size_t)(row0 + m) * IND + kb;
  const float* wpan = Wv + (size_t)col0 * IND;          // tile base: W_v[col0, 0]
  const unsigned lds[2] = { (unsigned)(size_t)&sB[0][0][0],
                            (unsigned)(size_t)&sB[1][0][0] };

  if (wave == 0) {
    tdm_load_2d_f32(lds[0], wpan, IND, OUTD, 96, 64, IND);
    __builtin_amdgcn_s_wait_tensorcnt(0);
  }
  __syncthreads();

  v8f c0 = {}, c1 = {}, c2 = {}, c3 = {};
  for (int k = 0; k < 4; ++k) {
    const int e0 = k * 96;
    if (wave == 0 && k + 1 < 4) {
      tdm_load_2d_f32(lds[(k + 1) & 1], wpan + (e0 + 96), IND, OUTD, 96, 64, IND);
    }
    const float (*buf)[96] = sB[k & 1];
#pragma unroll 4
    for (int e = 0; e < 96; e += 4) {
      v2f a  = *(const v2f*)(arow + e0 + e);
      v2f b0 = *(const v2f*)(&buf[m +  0][e + kb]);
      v2f b1 = *(const v2f*)(&buf[m + 16][e + kb]);
      v2f b2 = *(const v2f*)(&buf[m + 32][e + kb]);
      v2f b3 = *(const v2f*)(&buf[m + 48][e + kb]);
      c0 = __builtin_amdgcn_wmma_f32_16x16x4_f32(false, a, false, b0, (short)0, c0, false, false);
      c1 = __builtin_amdgcn_wmma_f32_16x16x4_f32(false, a, false, b1, (short)0, c1, false, false);
      c2 = __builtin_amdgcn_wmma_f32_16x16x4_f32(false, a, false, b2, (short)0, c2, false, false);
      c3 = __builtin_amdgcn_wmma_f32_16x16x4_f32(false, a, false, b3, (short)0, c3, false, false);
    }
    if (wave == 0) __builtin_amdgcn_s_wait_tensorcnt(0);
    __syncthreads();
  }

#pragma unroll
  for (int v = 0; v < 8; ++v) {
    float* orow = V + (size_t)(row0 + v + hi) * OUTD + col0 + m;
    orow[0]  = c0[v];
    orow[16] = c1[v];
    orow[32] = c2[v];
    orow[48] = c3[v];
  }
}

// ---------------------------------------------------------------------------
// Kernel 3: out[r, i, h*32+d] = sum_j attn[h, i, j] * V[r, j, h*32+d]
// Block = one (r, h): GEMM [384 x 384] x [384 x 32].  V head strip staged in
// LDS via TDM as 96x32 chunks, double buffered (no strided global loads).
// Each wave: 3 i-tiles x 2 d-tiles (48 VGPRs of accumulators).
// Grid: 256*12 = 3072 blocks of 256 threads.
// ---------------------------------------------------------------------------
__global__ void out_gemm_kernel(const float* __restrict__ attn,
                                const float* __restrict__ V,
                                float*       __restrict__ out) {
  __shared__ float sV[2][96][32];      // 24 KB double buffer

  const int tid  = threadIdx.x;
  const int wave = tid >> 5;
  const int lane = tid & 31;

  const int h = blockIdx.x % NH;
  const int r = blockIdx.x / NH;
  const int o0 = h * HD;

  const int m  = lane & 15;
  const int kb = (lane >> 4) << 1;
  const int hi = (lane >> 4) ? 8 : 0;

  const float* vstrip = V + (size_t)r * LL * OUTD + o0;   // V[r, 0, o0]
  const unsigned lds[2] = { (unsigned)(size_t)&sV[0][0][0],
                            (unsigned)(size_t)&sV[1][0][0] };

  // attn A rows for the 3 i-tiles of this wave
  const float* at0 = attn + ((size_t)h * LL + (wave * 48 +  0 + m)) * LL + kb;
  const float* at1 = attn + ((size_t)h * LL + (wave * 48 + 16 + m)) * LL + kb;
  const float* at2 = attn + ((size_t)h * LL + (wave * 48 + 32 + m)) * LL + kb;

  if (wave == 0) {
    tdm_load_2d_f32(lds[0], vstrip, OUTD, LL, 32, 96, OUTD);
    __builtin_amdgcn_s_wait_tensorcnt(0);
  }
  __syncthreads();

  v8f c00 = {}, c01 = {}, c10 = {}, c11 = {}, c20 = {}, c21 = {};
  for (int k = 0; k < 4; ++k) {
    const int j0 = k * 96;
    if (wave == 0 && k + 1 < 4) {
      tdm_load_2d_f32(lds[(k + 1) & 1], vstrip + (size_t)(j0 + 96) * OUTD,
                      OUTD, LL, 32, 96, OUTD);
    }
    const float (*buf)[32] = sV[k & 1];
#pragma unroll 4
    for (int j = 0; j < 96; j += 4) {
      v2f b0, b1;
      b0.x = buf[j + kb][m];      b0.y = buf[j + kb + 1][m];
      b1.x = buf[j + kb][m + 16]; b1.y = buf[j + kb + 1][m + 16];
      v2f a0 = *(const v2f*)(at0 + j0 + j);
      v2f a1 = *(const v2f*)(at1 + j0 + j);
      v2f a2 = *(const v2f*)(at2 + j0 + j);
      c00 = __builtin_amdgcn_wmma_f32_16x16x4_f32(false, a0, false, b0, (short)0, c00, false, false);
      c01 = __builtin_amdgcn_wmma_f32_16x16x4_f32(false, a0, false, b1, (short)0, c01, false, false);
      c10 = __builtin_amdgcn_wmma_f32_16x16x4_f32(false, a1, false, b0, (short)0, c10, false, false);
      c11 = __builtin_amdgcn_wmma_f32_16x16x4_f32(false, a1, false, b1, (short)0, c11, false, false);
      c20 = __builtin_amdgcn_wmma_f32_16x16x4_f32(false, a2, false, b0, (short)0, c20, false, false);
      c21 = __builtin_amdgcn_wmma_f32_16x16x4_f32(false, a2, false, b1, (short)0, c21, false, false);
    }
    if (wave == 0) __builtin_amdgcn_s_wait_tensorcnt(0);
    __syncthreads();
  }

#pragma unroll
  for (int v = 0; v < 8; ++v) {
    float* o0p = out + ((size_t)r * LL + (wave * 48 +  0 + v + hi)) * OUTD + o0 + m;
    float* o1p = out + ((size_t)r * LL + (wave * 48 + 16 + v + hi)) * OUTD + o0 + m;
    float* o2p = out + ((size_t)r * LL + (wave * 48 + 32 + v + hi)) * OUTD + o0 + m;
    o0p[0] = c00[v];  o0p[16] = c01[v];
    o1p[0] = c10[v];  o1p[16] = c11[v];
    o2p[0] = c20[v];  o2p[16] = c21[v];
  }
}

extern "C" void kernel_launch(void* const* d_in, const int* in_sizes, int n_in,
                              void* d_out, int out_size, void* d_ws, size_t ws_size,
                              hipStream_t stream) {
  const float* msa  = (const float*)d_in[0];  // [1,256,384,384]
  const float* pair = (const float*)d_in[1];  // [1,384,384,128]
  const float* Wl   = (const float*)d_in[2];  // [12,128]
  const float* Wv   = (const float*)d_in[3];  // [384,384]
  const int*   mask = (const int*)  d_in[4];  // [1,384]
  float*       out  = (float*)d_out;          // [1,256,384,384]

  float* attn = (float*)d_ws;                       // 12*384*384 floats = 7.08 MB
  float* V    = attn + (size_t)NH * LL * LL;        // 256*384*384 floats = 151 MB

  // Pass 1: logits + softmax -> attn
  attn_logits_softmax_kernel<<<LL, 384, 0, stream>>>(pair, Wl, mask, attn);

  // Pass 2: V = msa @ W_v^T  (TDM-staged W_v panels)
  v_gemm_kernel<<<(RR * LL / 128) * (OUTD / 64), 256, 0, stream>>>(msa, Wv, V);

  // Pass 3: out = attn @ V  (TDM-staged V head strips)
  out_gemm_kernel<<<RR * NH, 256, 0, stream>>>(attn, V, out);
}